// RelationAwareLinear_22892175687782
// MI455X (gfx1250) — compile-verified
//
#include <hip/hip_runtime.h>

// Problem constants (from reference): N=8192, R=16, IN_C=OUT_C=128
#define N_ROWS 8192
#define R_REL  16
#define IN_CH  128
#define OUT_CH 128
#define TILE_M 16
// sum_r ceil(c_r/16) <= (N + 15R)/16 = 527
#define MAX_TILES 528

// Workspace layout (int offsets)
#define OFF_COUNTS 0              // R ints
#define OFF_TOTAL  16             // 1 int
#define OFF_TREL   32             // MAX_TILES ints
#define OFF_TSTART (32 + MAX_TILES)
#define OFF_BUCKET 2048           // R * N ints

typedef __attribute__((ext_vector_type(2))) float v2f;
typedef __attribute__((ext_vector_type(8))) float v8f;

// ---------------- Pass 0: zero counters (must re-run every launch) ----------
__global__ void raw_init_kernel(int* __restrict__ ws) {
    const int t = threadIdx.x;
    if (t < R_REL) ws[OFF_COUNTS + t] = 0;
    if (t == 0)    ws[OFF_TOTAL] = 0;
}

// ---------------- Pass 1: bucket rows by relation ---------------------------
__global__ void raw_bucket_kernel(const int* __restrict__ relations,
                                  int* __restrict__ ws) {
    const int i = blockIdx.x * blockDim.x + threadIdx.x;
    if (i < N_ROWS) {
        const int r   = relations[i];
        const int pos = atomicAdd(&ws[OFF_COUNTS + r], 1);
        ws[OFF_BUCKET + r * N_ROWS + pos] = i;
    }
}

// ---------------- Pass 2: build tile descriptors ----------------------------
__global__ void raw_plan_kernel(int* __restrict__ ws) {
    if (threadIdx.x == 0 && blockIdx.x == 0) {
        int tnum = 0;
        for (int r = 0; r < R_REL; ++r) {
            const int c = ws[OFF_COUNTS + r];
            for (int s = 0; s < c; s += TILE_M) {
                ws[OFF_TREL + tnum]   = r;
                ws[OFF_TSTART + tnum] = s;
                ++tnum;
            }
        }
        ws[OFF_TOTAL] = tnum;
    }
}

// ---------------- Pass 3: grouped GEMM via V_WMMA_F32_16X16X4_F32 ----------
// Block: 256 threads = 8 wave32. One block = one 16-row tile (single relation).
// Wave w owns output columns [16w, 16w+16). K accumulated with 2 independent
// WMMA chains to hide the XDL RAW latency.
__global__ __launch_bounds__(256)
void raw_gemm_kernel(const float* __restrict__ x,
                     const float* __restrict__ w,
                     const float* __restrict__ bias,
                     const int* __restrict__ ws,
                     float* __restrict__ out) {
    const int total = ws[OFF_TOTAL];
    const int tile  = blockIdx.x;
    if (tile >= total) return;               // uniform across block

    const int rel    = ws[OFF_TREL + tile];
    const int bstart = ws[OFF_TSTART + tile];
    const int cnt    = ws[OFF_COUNTS + rel];
    const int* __restrict__ bucket = ws + OFF_BUCKET + rel * N_ROWS;

    __shared__ float xs[TILE_M][IN_CH + 4];  // +4 pad vs 64-bank LDS
    __shared__ int   rowIdx[TILE_M];

    const int t = threadIdx.x;
    {   // Gather 16 rows of x into LDS; zero-pad rows past bucket end.
        const int m  = t >> 4;               // tile row 0..15
        const int c0 = (t & 15) * 8;         // 8 floats per thread
        const int bi = bstart + m;
        int row = -1;
        if (bi < cnt) row = bucket[bi];
        if ((t & 15) == 0) rowIdx[m] = row;
        if (row >= 0) {
            const float4* src = (const float4*)(x + (size_t)row * IN_CH + c0);
            *(float4*)&xs[m][c0]     = src[0];
            *(float4*)&xs[m][c0 + 4] = src[1];
        } else {
            const float4 z = {0.f, 0.f, 0.f, 0.f};
            *(float4*)&xs[m][c0]     = z;
            *(float4*)&xs[m][c0 + 4] = z;
        }
    }
    __syncthreads();

    const int lane = t & 31;
    const int wav  = t >> 5;                 // 0..7
    const int n0   = wav * 16;               // this wave's output columns
    const int nl   = lane & 15;              // N within slab / A row M
    const int hi   = lane >> 4;              // lane half (ISA K-split)
    const int M    = lane & 15;

    // B base: w[rel][k][n0+nl], row-major [IN_CH][OUT_CH]
    const float* __restrict__ wr = w + (size_t)rel * IN_CH * OUT_CH + n0 + nl;

    v8f acc0 = {};
    v8f acc1 = {};
#pragma unroll
    for (int k = 0; k < IN_CH; k += 8) {
        // chain 0: K-slice [k, k+4)
        const int k0 = k + 2 * hi;
        v2f a0 = *(const v2f*)&xs[M][k0];    // {A[M][k0], A[M][k0+1]}
        v2f b0;
        b0.x = wr[(size_t)k0 * OUT_CH];      // B[k0][n]
        b0.y = wr[(size_t)(k0 + 1) * OUT_CH];
        acc0 = __builtin_amdgcn_wmma_f32_16x16x4_f32(
            false, a0, false, b0, (short)0, acc0, false, false);

        // chain 1: K-slice [k+4, k+8)
        const int k1 = k + 4 + 2 * hi;
        v2f a1 = *(const v2f*)&xs[M][k1];
        v2f b1;
        b1.x = wr[(size_t)k1 * OUT_CH];
        b1.y = wr[(size_t)(k1 + 1) * OUT_CH];
        acc1 = __builtin_amdgcn_wmma_f32_16x16x4_f32(
            false, a1, false, b1, (short)0, acc1, false, false);
    }

    const float bv = bias[rel * OUT_CH + n0 + nl];
#pragma unroll
    for (int v = 0; v < 8; ++v) {
        const int m = v + 8 * hi;            // C/D row: VGPR v, lane half hi
        if (bstart + m < cnt) {
            const int row = rowIdx[m];
            out[(size_t)row * OUT_CH + n0 + nl] = acc0[v] + acc1[v] + bv;
        }
    }
}

extern "C" void kernel_launch(void* const* d_in, const int* in_sizes, int n_in,
                              void* d_out, int out_size, void* d_ws, size_t ws_size,
                              hipStream_t stream) {
    const float* x         = (const float*)d_in[0];
    const int*   relations = (const int*)d_in[1];
    const float* w         = (const float*)d_in[2];
    const float* bias      = (const float*)d_in[3];
    float*       out       = (float*)d_out;
    int*         ws        = (int*)d_ws;

    raw_init_kernel<<<1, 32, 0, stream>>>(ws);
    raw_bucket_kernel<<<N_ROWS / 256, 256, 0, stream>>>(relations, ws);
    raw_plan_kernel<<<1, 32, 0, stream>>>(ws);
    raw_gemm_kernel<<<MAX_TILES, 256, 0, stream>>>(x, w, bias, ws, out);
}